// RBFBlock_4861902979088
// MI455X (gfx1250) — compile-verified
//
#include <hip/hip_runtime.h>

typedef float v2f __attribute__((ext_vector_type(2)));
typedef float v8f __attribute__((ext_vector_type(8)));

#define NXPTS 32768
#define NPTS  4096
#define MOUT  16
#define WAVES_PER_BLOCK 8
#define BLOCK_THREADS (WAVES_PER_BLOCK * 32)
#define ROWS_PER_BLOCK (WAVES_PER_BLOCK * 16)

// Fused RBF block: per wave, a 16-row x-tile; loop K=4096 in chunks of 4 using
// V_WMMA_F32_16X16X4_F32. A fragment accumulates a' = r2c*log2(r2c) (the
// constant 0.5*ln2 of tps(r)=r^2 ln r = 0.5*ln2 * r2 * log2(r2) is factored
// out of the linear contraction and applied once at the store). B fragment =
// coeffs[k][n] streamed from global, C accumulates fp32.
__global__ __launch_bounds__(BLOCK_THREADS) void rbf_fused_wmma(
    const float* __restrict__ x, const float* __restrict__ y,
    const float* __restrict__ shift, const float* __restrict__ scale,
    const float* __restrict__ coeffs, float* __restrict__ out)
{
    __shared__ float4 ys[NPTS];   // 64 KB: y padded to float4 for aligned b128 loads

    const int tid = threadIdx.x;
    for (int j = tid; j < NPTS; j += BLOCK_THREADS) {
        ys[j] = make_float4(y[3 * j + 0], y[3 * j + 1], y[3 * j + 2], 0.0f);
    }
    __syncthreads();

    const int lane   = tid & 31;
    const int wave   = tid >> 5;
    const int laneHi = lane >> 4;     // 0: K pair {0,1}, 1: K pair {2,3}
    const int m      = lane & 15;     // row (A) / col (B,C,D) within tile
    const int tile   = (blockIdx.x * WAVES_PER_BLOCK + wave) * 16;

    // A-fragment source: this lane's x row, kept in registers for the whole K loop
    const float* xr = x + (size_t)(tile + m) * 3;
    const float xa0 = xr[0], xa1 = xr[1], xa2 = xr[2];

    v8f c = {0.f, 0.f, 0.f, 0.f, 0.f, 0.f, 0.f, 0.f};

    const int jbase = laneHi * 2;                     // lane's K offset within chunk
    const float*  cb = coeffs + jbase * MOUT + m;     // coeffs[j][m], advances with K
    const float4* yb = ys + jbase;

    const float HLN2   = 0.34657359027997264f;  // 0.5 * ln(2), applied at epilogue
    const float R2_MIN = 1e-14f;                // (1e-7)^2

    #pragma unroll 4
    for (int k0 = 0; k0 < NPTS; k0 += 4) {
        const float4 p0 = yb[0];
        const float4 p1 = yb[1];
        const float  b0 = cb[0];
        const float  b1 = cb[MOUT];

        float dx = xa0 - p0.x, dy = xa1 - p0.y, dz = xa2 - p0.z;
        float r2 = fmaxf(dx * dx + dy * dy + dz * dz, R2_MIN);
        const float a0 = r2 * __builtin_amdgcn_logf(r2);   // r2c * log2(r2c)

        dx = xa0 - p1.x; dy = xa1 - p1.y; dz = xa2 - p1.z;
        r2 = fmaxf(dx * dx + dy * dy + dz * dz, R2_MIN);
        const float a1 = r2 * __builtin_amdgcn_logf(r2);

        const v2f a = {a0, a1};
        const v2f b = {b0, b1};
        // D = A(16x4) * B(4x16) + C, all fp32
        c = __builtin_amdgcn_wmma_f32_16x16x4_f32(
                /*neg_a=*/false, a, /*neg_b=*/false, b,
                /*c_mod=*/(short)0, c, /*reuse_a=*/false, /*reuse_b=*/false);

        yb += 4;
        cb += 4 * MOUT;
    }

    // Degree-1 polynomial tail: [1, xh0, xh1, xh2] @ coeffs[4096:4100], folded at store.
    const float s0 = shift[0], s1 = shift[1], s2 = shift[2];
    const float i0 = 1.0f / scale[0], i1 = 1.0f / scale[1], i2 = 1.0f / scale[2];
    const float* cp = coeffs + (size_t)NPTS * MOUT;
    const float q0 = cp[0 * MOUT + m];
    const float q1 = cp[1 * MOUT + m];
    const float q2 = cp[2 * MOUT + m];
    const float q3 = cp[3 * MOUT + m];

    #pragma unroll
    for (int i = 0; i < 8; ++i) {
        const int row = tile + i + laneHi * 8;        // C/D layout: VGPR i -> M=i (+8 hi half)
        const float* xq = x + (size_t)row * 3;
        const float xh0 = (xq[0] - s0) * i0;
        const float xh1 = (xq[1] - s1) * i1;
        const float xh2 = (xq[2] - s2) * i2;
        const float poly = q0 + xh0 * q1 + xh1 * q2 + xh2 * q3;
        out[(size_t)row * MOUT + m] = HLN2 * c[i] + poly;   // re-apply 0.5*ln2
    }
}

extern "C" void kernel_launch(void* const* d_in, const int* in_sizes, int n_in,
                              void* d_out, int out_size, void* d_ws, size_t ws_size,
                              hipStream_t stream) {
    const float* x      = (const float*)d_in[0];
    const float* y      = (const float*)d_in[1];
    const float* shift  = (const float*)d_in[2];
    const float* scale  = (const float*)d_in[3];
    const float* coeffs = (const float*)d_in[4];
    // d_in[5] = powers (int32): degree-1 monomials [1, x0, x1, x2] are hardcoded.
    float* out = (float*)d_out;
    (void)in_sizes; (void)n_in; (void)out_size; (void)d_ws; (void)ws_size;

    dim3 grid(NXPTS / ROWS_PER_BLOCK);   // 256 blocks x 256 threads (8 waves)
    rbf_fused_wmma<<<grid, BLOCK_THREADS, 0, stream>>>(x, y, shift, scale, coeffs, out);
}